// UltralyticsSegment_65773129171253
// MI455X (gfx1250) — compile-verified
//
#include <hip/hip_runtime.h>

typedef __attribute__((ext_vector_type(2))) float v2f;
typedef __attribute__((ext_vector_type(8))) float v8f;

#define B_    4
#define N_    8400
#define NM_   32
#define H_    160
#define W_    160
#define HW_   (H_*W_)          // 25600
#define D_    100
#define DP_   112              // D padded to 7*16 for WMMA row tiles
#define DT_   (DP_/16)         // 7 d-tiles
#define PT_   (HW_/16)         // 1600 pixel tiles
#define PG_   8                // pixel tiles per block (one per wave)
#define UPH_  (H_*4)
#define UPW_  (W_*4)

// ---------------------------------------------------------------------------
// Kernel 1: gather selected mask coefficients into padded A [B, 112, 32]
// ---------------------------------------------------------------------------
__global__ void gather_kernel(const float* __restrict__ mc,
                              const int* __restrict__ det,
                              float* __restrict__ Apad) {
  int tid = blockIdx.x * blockDim.x + threadIdx.x;
  if (tid >= B_ * DP_ * NM_) return;
  int n = tid & (NM_ - 1);
  int d = (tid >> 5) % DP_;
  int b = tid / (DP_ * NM_);
  float v = 0.0f;
  if (d < D_) {
    int idx = det[b * D_ + d];
    v = mc[((size_t)b * N_ + (size_t)idx) * NM_ + n];
  }
  Apad[tid] = v;
}

// ---------------------------------------------------------------------------
// Kernel 2: f32 WMMA GEMM (16x16 tile per wave, K=32 as 8x 16x16x4_f32)
//           fused with fast sigmoid (v_rcp_f32); writes padded low-res field
//           Slow[B, 112, 160*160] unconditionally (branch-free epilogue).
// ---------------------------------------------------------------------------
__global__ void __launch_bounds__(256)
wmma_sigmoid_kernel(const float* __restrict__ Apad,
                    const float* __restrict__ proto,
                    float* __restrict__ Slow) {
  const int wave  = threadIdx.x >> 5;
  const int lane  = threadIdx.x & 31;
  const int blk   = blockIdx.x;
  const int pg    = blk % (PT_ / PG_);
  const int dt    = (blk / (PT_ / PG_)) % DT_;
  const int b     = blk / ((PT_ / PG_) * DT_);
  const int ptile = pg * PG_ + wave;
  const int p0    = ptile * 16;

  const int n     = lane & 15;   // row of A tile / column of B,C tiles
  const int khalf = lane >> 4;   // 0: K+0/K+1  1: K+2/K+3 (A & B halves)

  // A-matrix 16x4 f32 layout: lanes 0-15 hold M=0..15, VGPR0={K0|K2}, VGPR1={K1|K3}
  const float* Arow = Apad + ((size_t)(b * DP_ + dt * 16 + n) * NM_);
  const float* Bb   = proto + (size_t)b * NM_ * HW_;

  v8f acc = {};
#pragma unroll
  for (int k = 0; k < NM_; k += 4) {
    v2f a, bm;
    a.x  = Arow[k + 2 * khalf];
    a.y  = Arow[k + 2 * khalf + 1];
    // B-matrix 4x16 f32: lane's N = lane&15; VGPR0={K0|K2}, VGPR1={K1|K3}
    bm.x = Bb[(size_t)(k + 2 * khalf)     * HW_ + p0 + n];
    bm.y = Bb[(size_t)(k + 2 * khalf + 1) * HW_ + p0 + n];
    acc = __builtin_amdgcn_wmma_f32_16x16x4_f32(
        /*neg_a=*/false, a, /*neg_b=*/false, bm,
        /*c_mod=*/(short)0, acc, /*reuse_a=*/false, /*reuse_b=*/false);
  }

  // C/D layout: VGPR v -> M = v + 8*khalf, N = lane&15.
  // Store all 112 padded rows: no exec-mask divergence in the epilogue.
  float* outBase = Slow + ((size_t)(b * DP_ + dt * 16 + 8 * khalf)) * HW_ + p0 + n;
#pragma unroll
  for (int v = 0; v < 8; ++v) {
    // sigmoid via single v_exp_f32 + v_rcp_f32 (~1 ulp), no div fixup chain
    float s = __builtin_amdgcn_rcpf(1.0f + __expf(-acc[v]));
    outBase[(size_t)v * HW_] = s;
  }
}

// ---------------------------------------------------------------------------
// Kernel 3: half-pixel bilinear x4 upsample + threshold (>0.5 -> 1.0f)
//           one thread -> one 4-wide output x-cell, float4 coalesced store.
// src = o/4 - 0.375; per sub-pixel r: weight of "upper" neighbor is
// r=0: pair (i-1,i)  w1=0.625 ; r=1: (i-1,i) w1=0.875
// r=2: pair (i,i+1)  w1=0.125 ; r=3: (i,i+1) w1=0.375   (edges clamped)
// ---------------------------------------------------------------------------
__global__ void __launch_bounds__(256)
upsample_threshold_kernel(const float* __restrict__ S,
                          float* __restrict__ out) {
  int tid = blockIdx.x * blockDim.x + threadIdx.x;  // B_*D_*UPH_*W_ threads
  int ix  = tid % W_;
  int oy  = (tid / W_) % UPH_;
  int img = tid / (W_ * UPH_);      // [0, B_*D_)
  int b   = img / D_;
  int d   = img % D_;

  int iy = oy >> 2, ry = oy & 3;
  int y0 = iy + ((ry < 2) ? -1 : 0);
  int y1 = y0 + 1;
  float wy1 = (ry == 0) ? 0.625f : (ry == 1) ? 0.875f
            : (ry == 2) ? 0.125f : 0.375f;
  y0 = (y0 < 0) ? 0 : y0;
  y1 = (y1 > H_ - 1) ? H_ - 1 : y1;

  const float* base = S + (size_t)(b * DP_ + d) * HW_;   // padded layout
  const float* r0 = base + y0 * W_;
  const float* r1 = base + y1 * W_;
  int xm = (ix > 0) ? ix - 1 : 0;
  int xp = (ix < W_ - 1) ? ix + 1 : W_ - 1;

  float s0m = r0[xm], s00 = r0[ix], s0p = r0[xp];
  float s1m = r1[xm], s10 = r1[ix], s1p = r1[xp];

  // vertical blend
  float vm = s0m + wy1 * (s1m - s0m);
  float v0 = s00 + wy1 * (s10 - s00);
  float vp = s0p + wy1 * (s1p - s0p);

  float4 o;
  o.x = (0.375f * vm + 0.625f * v0) > 0.5f ? 1.0f : 0.0f;
  o.y = (0.125f * vm + 0.875f * v0) > 0.5f ? 1.0f : 0.0f;
  o.z = (0.875f * v0 + 0.125f * vp) > 0.5f ? 1.0f : 0.0f;
  o.w = (0.625f * v0 + 0.375f * vp) > 0.5f ? 1.0f : 0.0f;

  // out offset = img*UPH_*UPW_ + oy*UPW_ + ix*4 == 4*tid
  ((float4*)out)[(size_t)tid] = o;
}

// ---------------------------------------------------------------------------
extern "C" void kernel_launch(void* const* d_in, const int* in_sizes, int n_in,
                              void* d_out, int out_size, void* d_ws, size_t ws_size,
                              hipStream_t stream) {
  const float* mc    = (const float*)d_in[0];
  const float* proto = (const float*)d_in[1];
  const int*   det   = (const int*)d_in[2];
  float* out  = (float*)d_out;

  float* Apad = (float*)d_ws;                         // B*112*32 floats (56 KB)
  float* Slow = Apad + (size_t)B_ * DP_ * NM_;        // B*112*25600 floats (~45.9 MB, L2-resident)

  // 1) gather (B*112*32 = 14336 elements)
  {
    int total = B_ * DP_ * NM_;
    gather_kernel<<<(total + 255) / 256, 256, 0, stream>>>(mc, det, Apad);
  }

  // 2) WMMA GEMM + sigmoid: 4 batches * 7 d-tiles * 200 pixel-tile groups
  {
    int blocks = B_ * DT_ * (PT_ / PG_);              // 5600 blocks, 8 waves each
    wmma_sigmoid_kernel<<<blocks, 256, 0, stream>>>(Apad, proto, Slow);
  }

  // 3) upsample + threshold: B*D*640*160 threads = 40.96M (exact multiple of 256)
  {
    int total = B_ * D_ * UPH_ * W_;
    upsample_threshold_kernel<<<total / 256, 256, 0, stream>>>(Slow, out);
  }
}